// TropicalLinearLayer_20332375179466
// MI455X (gfx1250) — compile-verified
//
#include <hip/hip_runtime.h>

#ifndef __has_builtin
#define __has_builtin(x) 0
#endif

#define TROPICAL_ZERO (-1e38f)
#define K_DIM   256   // IN
#define N_DIM   256   // OUT
#define BM      64    // batch-rows per block
#define BN      64    // out-cols per block
#define BK      32    // k chunk
#define LDW     36    // BK + 4 pad floats (keeps 16B alignment, spreads banks)
#define NCHUNK  (K_DIM / BK)
#define OPS_PER_CHUNK 4   // async b128 ops issued per thread per chunk

// CDNA5 async global->LDS DMA path via inline asm (portable across toolchains;
// bypasses the clang builtin whose params use CUDA language-AS sugar).
#if defined(__HIP_DEVICE_COMPILE__) && defined(__gfx1250__)
#define HAVE_ASYNC 1
#else
#define HAVE_ASYNC 0
#endif

__device__ __forceinline__ void copy_b128_g2lds(const float* g, float* l) {
#if HAVE_ASYNC
  // Low 32 bits of a flat pointer into __shared__ are the wave-relative LDS
  // byte address (flat-LDS aperture keeps the offset in addr[31:0]).
  unsigned lds_addr = (unsigned)(unsigned long long)l;
  asm volatile("global_load_async_to_lds_b128 %0, %1, off"
               :
               : "v"(lds_addr), "v"(g)
               : "memory");
#else
  *reinterpret_cast<float4*>(l) = *reinterpret_cast<const float4*>(g);
#endif
}

__device__ __forceinline__ void wait_async_prev_chunk() {
#if HAVE_ASYNC
#if __has_builtin(__builtin_amdgcn_s_wait_asynccnt)
  __builtin_amdgcn_s_wait_asynccnt(OPS_PER_CHUNK);  // next chunk still in flight
#else
  asm volatile("s_wait_asynccnt 0x4" ::: "memory");
#endif
#endif
}

__device__ __forceinline__ void wait_async_all() {
#if HAVE_ASYNC
#if __has_builtin(__builtin_amdgcn_s_wait_asynccnt)
  __builtin_amdgcn_s_wait_asynccnt(0);
#else
  asm volatile("s_wait_asynccnt 0x0" ::: "memory");
#endif
#endif
}

__global__ __launch_bounds__(256) void tropical_maxplus_kernel(
    const float* __restrict__ x, const float* __restrict__ w,
    const float* __restrict__ bias, float* __restrict__ out) {
  // double-buffered tiles: x rows [BM][k], w rows [BN][k]
  __shared__ float xs[2][BM][LDW];
  __shared__ float ws[2][BN][LDW];

  const int tid = threadIdx.x;
  const int tx  = tid & 15;   // col group 0..15  -> cols tx*4..tx*4+3
  const int ty  = tid >> 4;   // row group 0..15  -> rows ty*4..ty*4+3

  const long rowBase = (long)blockIdx.x * BM;
  const int  colBase = blockIdx.y * BN;

  const float* xg = x + rowBase * K_DIM;          // 64 rows of x
  const float* wg = w + (long)colBase * K_DIM;    // 64 rows of w (w is [OUT][IN])

  // loader mapping: lanes 0..7 cover one 128B row segment (coalesced b128)
  const int lk = (tid & 7) * 4;   // k offset within chunk: 0..28
  const int lr = tid >> 3;        // row 0..31 (plus +32 second pass)

  float acc[4][4];
#pragma unroll
  for (int i = 0; i < 4; ++i)
#pragma unroll
    for (int j = 0; j < 4; ++j) acc[i][j] = -__builtin_inff();

  auto stage = [&](int buf, int kb) {
    copy_b128_g2lds(xg + (long)(lr     ) * K_DIM + kb + lk, &xs[buf][lr     ][lk]);
    copy_b128_g2lds(xg + (long)(lr + 32) * K_DIM + kb + lk, &xs[buf][lr + 32][lk]);
    copy_b128_g2lds(wg + (long)(lr     ) * K_DIM + kb + lk, &ws[buf][lr     ][lk]);
    copy_b128_g2lds(wg + (long)(lr + 32) * K_DIM + kb + lk, &ws[buf][lr + 32][lk]);
  };

  stage(0, 0);

#pragma unroll 1
  for (int kc = 0; kc < NCHUNK; ++kc) {
    const int cur = kc & 1;
    if (kc + 1 < NCHUNK) {
      stage(cur ^ 1, (kc + 1) * BK);  // issue next chunk first (in-order cnt)
      wait_async_prev_chunk();        // then wait only for current chunk
    } else {
      wait_async_all();
    }
    __syncthreads();

#pragma unroll
    for (int kk = 0; kk < BK; kk += 4) {
      float4 a[4], b[4];
#pragma unroll
      for (int i = 0; i < 4; ++i)
        a[i] = *reinterpret_cast<const float4*>(&xs[cur][ty * 4 + i][kk]);
#pragma unroll
      for (int j = 0; j < 4; ++j)
        b[j] = *reinterpret_cast<const float4*>(&ws[cur][tx * 4 + j][kk]);
#pragma unroll
      for (int i = 0; i < 4; ++i) {
#pragma unroll
        for (int j = 0; j < 4; ++j) {
          // structured so the backend can form v_pk_add_f32 / v_max3_num_f32
          float p0 = a[i].x + b[j].x;
          float p1 = a[i].y + b[j].y;
          float p2 = a[i].z + b[j].z;
          float p3 = a[i].w + b[j].w;
          acc[i][j] = fmaxf(acc[i][j], fmaxf(fmaxf(p0, p1), fmaxf(p2, p3)));
        }
      }
    }
    __syncthreads();
  }

  // epilogue: mask + bias, vectorized stores
  const float4 bv = *reinterpret_cast<const float4*>(&bias[colBase + tx * 4]);
  const float bj[4] = {bv.x, bv.y, bv.z, bv.w};
#pragma unroll
  for (int i = 0; i < 4; ++i) {
    const long row = rowBase + ty * 4 + i;
    float4 o;
    o.x = (acc[i][0] > TROPICAL_ZERO) ? (acc[i][0] + bj[0]) : TROPICAL_ZERO;
    o.y = (acc[i][1] > TROPICAL_ZERO) ? (acc[i][1] + bj[1]) : TROPICAL_ZERO;
    o.z = (acc[i][2] > TROPICAL_ZERO) ? (acc[i][2] + bj[2]) : TROPICAL_ZERO;
    o.w = (acc[i][3] > TROPICAL_ZERO) ? (acc[i][3] + bj[3]) : TROPICAL_ZERO;
    *reinterpret_cast<float4*>(&out[row * N_DIM + colBase + tx * 4]) = o;
  }
}

extern "C" void kernel_launch(void* const* d_in, const int* in_sizes, int n_in,
                              void* d_out, int out_size, void* d_ws, size_t ws_size,
                              hipStream_t stream) {
  const float* x    = (const float*)d_in[0];
  const float* w    = (const float*)d_in[1];
  const float* bias = (const float*)d_in[2];
  float* out        = (float*)d_out;

  const int Bdim = in_sizes[0] / K_DIM;  // 4096
  dim3 grid(Bdim / BM, N_DIM / BN);      // (64, 4)
  tropical_maxplus_kernel<<<grid, dim3(256), 0, stream>>>(x, w, bias, out);

  (void)n_in; (void)out_size; (void)d_ws; (void)ws_size;
}